// BezierHCPathOptimizerJointStart_63032940036317
// MI455X (gfx1250) — compile-verified
//
#include <hip/hip_runtime.h>
#include <math.h>

typedef __attribute__((ext_vector_type(2))) float v2f;
typedef __attribute__((ext_vector_type(8))) float v8f;

#define NBLK 512
#define BS   256

// d_ws float layout:
//  [0..63]    P_ctrl (4 rows x 16 reals)
//  [64..111]  Q = 3*(Pc[k+1]-Pc[k])   (3 x 16)
//  [112..143] R = 2*(Q[k+1]-Q[k])     (2 x 16)
//  [160..]    per-block partials, 4 floats each
#define WS_PC   0
#define WS_Q    64
#define WS_R    112
#define WS_PART 160

__device__ __forceinline__ float2 cmulf(float2 a, float2 b) {
  return make_float2(a.x*b.x - a.y*b.y, a.x*b.y + a.y*b.x);
}
__device__ __forceinline__ float2 csubf(float2 a, float2 b) {
  return make_float2(a.x - b.x, a.y - b.y);
}
__device__ __forceinline__ float logaddexpf_(float a, float b) {
  float mx = fmaxf(a, b), mn = fminf(a, b);
  return mx + log1pf(expf(mn - mx));
}

// ---------------- setup: build P_ctrl / Q / R from the 5 tiny inputs ----------
__global__ void bezier_setup(const float* __restrict__ Pd, const float* __restrict__ c_re,
                             const float* __restrict__ c_im, const float* __restrict__ log_rho,
                             const float* __restrict__ theta, const float* __restrict__ P_mid,
                             float* __restrict__ wsf)
{
  if (threadIdx.x != 0 || blockIdx.x != 0) return;
  const float comb[9] = {1.f, 8.f, 28.f, 56.f, 70.f, 56.f, 28.f, 8.f, 1.f};
  float cre = c_re[0], cim = c_im[0];
  float rho = expf(log_rho[0]) + 1e-8f;
  float th  = theta[0];
  float sx = rho * cosf(th), sy = rho * sinf(th);
  float s2x = sx*sx - sy*sy,      s2y = 2.f*sx*sy;
  float s4x = s2x*s2x - s2y*s2y,  s4y = 2.f*s2x*s2y;
  float s8x = s4x*s4x - s4y*s4y,  s8y = 2.f*s4x*s4y;
  float fx[9], fy[9];
  float px = 1.f, py = 0.f;                    // (-c)^j
  for (int j = 0; j < 9; ++j) {
    fx[j] = comb[j]*px; fy[j] = comb[j]*py;
    float nx = px*(-cre) - py*(-cim);
    float ny = px*(-cim) + py*(-cre);
    px = nx; py = ny;
  }
  fx[8] -= s8x; fy[8] -= s8y;                  // full[-1] += -s^n
  float Pc[4][16];
  for (int j = 0; j < 8; ++j) { Pc[0][2*j] = fx[j+1]; Pc[0][2*j+1] = fy[j+1]; }
  for (int i = 0; i < 16; ++i) { Pc[1][i] = P_mid[i]; Pc[2][i] = P_mid[16+i]; Pc[3][i] = Pd[i]; }
  float Q[3][16], R[2][16];
  for (int k = 0; k < 3; ++k) for (int i = 0; i < 16; ++i) Q[k][i] = 3.f*(Pc[k+1][i] - Pc[k][i]);
  for (int k = 0; k < 2; ++k) for (int i = 0; i < 16; ++i) R[k][i] = 2.f*(Q[k+1][i] - Q[k][i]);
  for (int k = 0; k < 4; ++k) for (int i = 0; i < 16; ++i) wsf[WS_PC + k*16 + i] = Pc[k][i];
  for (int k = 0; k < 3; ++k) for (int i = 0; i < 16; ++i) wsf[WS_Q  + k*16 + i] = Q[k][i];
  for (int k = 0; k < 2; ++k) for (int i = 0; i < 16; ++i) wsf[WS_R  + k*16 + i] = R[k][i];
}

// ------------- one WMMA stage: (16 samples x 4 Bernstein) @ (4 x 16 ctrl) -------------
// A 16x4 f32 layout: lanes 0-15 row M=lane, VGPR0={K0|K2}, VGPR1={K1|K3}.
// C/D 16x16 f32 layout: VGPR v -> rows {v, v+8} across lane halves, col = lane&15.
template<int STAGE>
__device__ __forceinline__ void wmma_stage(float (&sbuf)[8][32][16], int w, int lane,
                                           int mA, int Mm1, float invM1, v2f bfrag)
{
  const int hi  = lane >> 4;
  const int col = lane & 15;
  #pragma unroll
  for (int h = 0; h < 2; ++h) {
    int msamp = mA + h*16 + col;
    int mc = msamp < Mm1 ? msamp : Mm1;
    float t = (float)mc * invM1;
    float u = 1.0f - t;
    v2f a;
    if (STAGE == 0) {            // Bernstein degree 3
      float b0 = u*u*u, b1 = 3.f*t*u*u, b2 = 3.f*t*t*u, b3 = t*t*t;
      a.x = hi ? b2 : b0; a.y = hi ? b3 : b1;
    } else if (STAGE == 1) {     // degree 2 (K=3 weight zero)
      float c0 = u*u, c1 = 2.f*t*u, c2 = t*t;
      a.x = hi ? c2 : c0; a.y = hi ? 0.f : c1;
    } else {                     // degree 1 (K=2,3 weights zero)
      a.x = hi ? 0.f : u; a.y = hi ? 0.f : t;
    }
    v8f acc = {};
    acc = __builtin_amdgcn_wmma_f32_16x16x4_f32(false, a, false, bfrag,
                                                (short)0, acc, false, false);
    #pragma unroll
    for (int v = 0; v < 8; ++v)
      sbuf[w][h*16 + v + 8*hi][col] = acc[v];
  }
}

// ---------------- main: WMMA curve eval + per-lane O(n^2) resultant ----------------
__global__ __launch_bounds__(BS) void bezier_main(const float* __restrict__ wsf,
                                                  const int* __restrict__ Mp,
                                                  float* __restrict__ partials)
{
  __shared__ float sbuf[8][32][16];
  __shared__ float red[3][BS];
  const int M    = Mp[0];
  const int Mm1  = (M > 1) ? (M - 1) : 1;
  const float invM1 = 1.0f / (float)Mm1;
  const int tid  = threadIdx.x;
  const int w    = tid >> 5, lane = tid & 31;
  const int hi   = lane >> 4, col = lane & 15;

  // B fragments (KxN = 4x16): VGPR0 = rows {K0|K2}, VGPR1 = rows {K1|K3}
  v2f bT, bTp, bTpp;
  bT.x   = wsf[WS_PC + (hi ? 2 : 0)*16 + col];
  bT.y   = wsf[WS_PC + (hi ? 3 : 1)*16 + col];
  bTp.x  = hi ? wsf[WS_Q + 2*16 + col] : wsf[WS_Q + 0*16 + col];
  bTp.y  = hi ? 0.f : wsf[WS_Q + 1*16 + col];
  bTpp.x = hi ? 0.f : wsf[WS_R + 0*16 + col];
  bTpp.y = hi ? 0.f : wsf[WS_R + 1*16 + col];

  float acc_sw = 0.f, acc_s2 = 0.f, acc_a2 = 0.f;

  for (int base = blockIdx.x * BS; base < M; base += gridDim.x * BS) {
    const int mA = base + w*32;

    wmma_stage<0>(sbuf, w, lane, mA, Mm1, invM1, bT);      // curve point -> coeffs
    __syncthreads();
    float Tv[16];
    #pragma unroll
    for (int i = 0; i < 16; ++i) Tv[i] = sbuf[w][lane][i];
    __syncthreads();

    wmma_stage<1>(sbuf, w, lane, mA, Mm1, invM1, bTp);     // 1st derivative
    __syncthreads();
    float s2 = 0.f;
    #pragma unroll
    for (int i = 0; i < 16; ++i) { float x = sbuf[w][lane][i]; s2 += x*x; }
    __syncthreads();

    wmma_stage<2>(sbuf, w, lane, mA, Mm1, invM1, bTpp);    // 2nd derivative
    __syncthreads();
    float a2s = 0.f;
    #pragma unroll
    for (int i = 0; i < 16; ++i) { float x = sbuf[w][lane][i]; a2s += x*x; }
    __syncthreads();

    // p(z) = z^8 + a0 z^7 + ... + a7 ; g = p' ; log|Res(p,p')| == log|det Sylvester|
    float2 f[9], g[9];
    f[8] = make_float2(1.f, 0.f);
    #pragma unroll
    for (int d = 0; d < 8; ++d) f[7-d] = make_float2(Tv[2*d], Tv[2*d+1]);
    #pragma unroll
    for (int k = 0; k < 8; ++k) g[k] = make_float2((float)(k+1)*f[k+1].x, (float)(k+1)*f[k+1].y);

    float logR = 0.f;
    #pragma unroll
    for (int deg = 8; deg >= 2; --deg) {             // f deg=deg, g deg=deg-1, r deg=deg-2
      float2 lg = g[deg-1];
      float d2  = fmaxf(lg.x*lg.x + lg.y*lg.y, 1e-37f);
      logR += logf(d2);                              // (deg f - deg r)=2 -> 2*log|lc| = log(|lc|^2)
      float inv = 1.0f / d2;
      float2 fn = f[deg];
      float2 q1 = make_float2((fn.x*lg.x + fn.y*lg.y)*inv, (fn.y*lg.x - fn.x*lg.y)*inv);
      float2 tp[9];
      tp[0] = f[0];
      #pragma unroll
      for (int i = 1; i <= 8; ++i) if (i <= deg-1) tp[i] = csubf(f[i], cmulf(q1, g[i-1]));
      float2 tl = tp[deg-1];
      float2 q0 = make_float2((tl.x*lg.x + tl.y*lg.y)*inv, (tl.y*lg.x - tl.x*lg.y)*inv);
      float2 r[9];
      #pragma unroll
      for (int i = 0; i <= 8; ++i) if (i <= deg-2) r[i] = csubf(tp[i], cmulf(q0, g[i]));
      #pragma unroll
      for (int i = 0; i <= 8; ++i) if (i <= deg-1) f[i] = g[i];
      #pragma unroll
      for (int i = 0; i <= 8; ++i) if (i <= deg-2) g[i] = r[i];
    }
    // terminal: Res(f_deg1, const) = const^1
    logR += 0.5f * logf(fmaxf(g[0].x*g[0].x + g[0].y*g[0].y, 1e-37f));

    float disc  = logaddexpf_(logR, -69.07755278982137f);          // log(1e-30)
    float lsoft = 0.5f * logaddexpf_(2.f*disc, -27.63102111592855f); // 2*log(1e-6)
    float lse   = logaddexpf_(lsoft, -13.815510557964274f);        // log(1e-6)
    float wgt   = expf(-lse * 0.125f);                             // exp(-lse/n), n=8
    float speed = sqrtf(s2);

    if (base + tid < M) {
      acc_sw += speed * wgt;
      acc_s2 += s2;
      acc_a2 += a2s;
    }
  }

  red[0][tid] = acc_sw; red[1][tid] = acc_s2; red[2][tid] = acc_a2;
  __syncthreads();
  for (int s = BS/2; s > 0; s >>= 1) {
    if (tid < s) {
      red[0][tid] += red[0][tid+s];
      red[1][tid] += red[1][tid+s];
      red[2][tid] += red[2][tid+s];
    }
    __syncthreads();
  }
  if (tid == 0) {
    partials[blockIdx.x*4 + 0] = red[0][0];
    partials[blockIdx.x*4 + 1] = red[1][0];
    partials[blockIdx.x*4 + 2] = red[2][0];
  }
}

// ---------------- final deterministic reduction ----------------
__global__ __launch_bounds__(BS) void bezier_final(const float* __restrict__ partials,
                                                   const int* __restrict__ Mp,
                                                   float* __restrict__ out, int nblocks)
{
  __shared__ float red[3][BS];
  int tid = threadIdx.x;
  float s0 = 0.f, s1 = 0.f, s2 = 0.f;
  for (int b = tid; b < nblocks; b += BS) {
    s0 += partials[b*4+0]; s1 += partials[b*4+1]; s2 += partials[b*4+2];
  }
  red[0][tid] = s0; red[1][tid] = s1; red[2][tid] = s2;
  __syncthreads();
  for (int s = BS/2; s > 0; s >>= 1) {
    if (tid < s) {
      red[0][tid] += red[0][tid+s];
      red[1][tid] += red[1][tid+s];
      red[2][tid] += red[2][tid+s];
    }
    __syncthreads();
  }
  if (tid == 0) {
    float Mf  = (float)Mp[0];
    float Lcl = red[0][0] / Mf;
    float Ld1 = sqrtf(red[1][0] / Mf);
    float Ld2 = sqrtf(red[2][0] / Mf);
    out[0] = Lcl + 0.1f*Ld1 + 0.01f*Ld2;
  }
}

extern "C" void kernel_launch(void* const* d_in, const int* in_sizes, int n_in,
                              void* d_out, int out_size, void* d_ws, size_t ws_size,
                              hipStream_t stream)
{
  (void)in_sizes; (void)n_in; (void)out_size; (void)ws_size;
  const float* Pd      = (const float*)d_in[0];
  const float* c_re    = (const float*)d_in[1];
  const float* c_im    = (const float*)d_in[2];
  const float* log_rho = (const float*)d_in[3];
  const float* theta   = (const float*)d_in[4];
  const float* P_mid   = (const float*)d_in[5];
  const int*   Mp      = (const int*)d_in[6];

  float* wsf      = (float*)d_ws;
  float* partials = wsf + WS_PART;

  bezier_setup<<<1, 32, 0, stream>>>(Pd, c_re, c_im, log_rho, theta, P_mid, wsf);
  bezier_main<<<NBLK, BS, 0, stream>>>(wsf, Mp, partials);
  bezier_final<<<1, BS, 0, stream>>>(partials, Mp, (float*)d_out, NBLK);
}